// BetaPerm_26929444946081
// MI455X (gfx1250) — compile-verified
//
#include <hip/hip_runtime.h>
#include <hip/hip_bf16.h>
#include <math.h>
#include <stdint.h>

// ---------------- problem constants (match reference) ----------------
#define NS   2000    // samples
#define KC   10      // covariates
#define LV   10000   // variants
#define RNUM 200     // permutations
#define RP   208     // permutations padded to multiple of 16 for WMMA
#define DOF  1989    // N - K - 1
#define MAXIT_BETA 100
#define TOLF 1e-3f

typedef __attribute__((ext_vector_type(2))) float v2f;
typedef __attribute__((ext_vector_type(8))) float v8f;
typedef __attribute__((ext_vector_type(4))) unsigned int v4u;
typedef __attribute__((ext_vector_type(8))) int v8i;
typedef __attribute__((ext_vector_type(4))) int v4i;

// =====================================================================
// Kernel 1: XtX (10x10) + Gauss-Jordan inverse, all in LDS. One block.
// =====================================================================
__global__ void __launch_bounds__(128) xtx_inv_kernel(const float* __restrict__ X,
                                                      float* __restrict__ XtXinv) {
    __shared__ float s_xtx[KC * KC];
    __shared__ float s_aug[KC][2 * KC];
    int t = threadIdx.x;
    if (t < KC * KC) {
        int i = t / KC, j = t % KC;
        float s = 0.f;
        for (int n = 0; n < NS; ++n)
            s += X[(size_t)n * KC + i] * X[(size_t)n * KC + j];
        s_xtx[t] = s;
    }
    __syncthreads();
    if (t == 0) {
        for (int i = 0; i < KC; ++i)
            for (int j = 0; j < KC; ++j) {
                s_aug[i][j] = s_xtx[i * KC + j];
                s_aug[i][KC + j] = (i == j) ? 1.f : 0.f;
            }
        for (int c = 0; c < KC; ++c) {
            float ip = 1.f / s_aug[c][c];
            for (int j = 0; j < 2 * KC; ++j) s_aug[c][j] *= ip;
            for (int r = 0; r < KC; ++r) {
                if (r == c) continue;
                float f = s_aug[r][c];
                for (int j = 0; j < 2 * KC; ++j) s_aug[r][j] -= f * s_aug[c][j];
            }
        }
        for (int i = 0; i < KC; ++i)
            for (int j = 0; j < KC; ++j)
                XtXinv[i * KC + j] = s_aug[i][KC + j];
    }
}

// =====================================================================
// Kernel 2: per-variant residualization (fused X^T G, solve, project)
//   Gt[:,j] = G[:,j] - X (XtX)^-1 X^T G[:,j];  gg[j] = ||Gt[:,j]||^2
// =====================================================================
__global__ void __launch_bounds__(256) residualize_kernel(const float* __restrict__ X,
                                                          const float* __restrict__ G,
                                                          const float* __restrict__ XtXinv,
                                                          float* __restrict__ Gt,
                                                          float* __restrict__ gg) {
    int j = blockIdx.x * blockDim.x + threadIdx.x;
    if (j >= LV) return;
    float a[KC];
#pragma unroll
    for (int k = 0; k < KC; ++k) a[k] = 0.f;
    for (int n = 0; n < NS; ++n) {
        float g = G[(size_t)n * LV + j];
        __builtin_prefetch(&G[(size_t)(n + 16) * LV + j], 0, 1);
#pragma unroll
        for (int k = 0; k < KC; ++k) a[k] += X[(size_t)n * KC + k] * g;
    }
    float b[KC];
#pragma unroll
    for (int k = 0; k < KC; ++k) {
        float s = 0.f;
#pragma unroll
        for (int i = 0; i < KC; ++i) s += XtXinv[k * KC + i] * a[i];
        b[k] = s;
    }
    float acc = 0.f;
    for (int n = 0; n < NS; ++n) {
        float proj = 0.f;
#pragma unroll
        for (int k = 0; k < KC; ++k) proj += X[(size_t)n * KC + k] * b[k];
        float gt = G[(size_t)n * LV + j] - proj;
        Gt[(size_t)n * LV + j] = gt;
        acc += gt * gt;
    }
    gg[j] = acc;
}

// =====================================================================
// Kernel 3: 200 permutations (inside-out Fisher-Yates, 1 thread each),
// offset subtraction, and residual sum-of-squares yty[r].
// =====================================================================
__global__ void __launch_bounds__(256) perm_kernel(const float* __restrict__ y,
                                                   const float* __restrict__ offset_eta,
                                                   const float* __restrict__ X,
                                                   const float* __restrict__ XtXinv,
                                                   const int* __restrict__ seed,
                                                   float* __restrict__ Yp,
                                                   float* __restrict__ yty) {
    int r = threadIdx.x;
    if (r >= RP) return;
    if (r >= RNUM) {                 // zero pad columns
        for (int n = 0; n < NS; ++n) Yp[(size_t)n * RP + r] = 0.f;
        yty[r] = 1.f;
        return;
    }
    unsigned s = (unsigned)seed[0] * 0x9E3779B1u + (unsigned)(r + 1) * 0x85EBCA6Bu + 0x6A09E667u;
    for (int i = 0; i < NS; ++i) {
        s ^= s << 13; s ^= s >> 17; s ^= s << 5;
        unsigned jr = s % (unsigned)(i + 1);
        if (jr != (unsigned)i) Yp[(size_t)i * RP + r] = Yp[(size_t)jr * RP + r];
        Yp[(size_t)jr * RP + r] = y[i];
    }
    float s2 = 0.f, xty[KC];
#pragma unroll
    for (int k = 0; k < KC; ++k) xty[k] = 0.f;
    for (int n = 0; n < NS; ++n) {
        float v = Yp[(size_t)n * RP + r] - offset_eta[n];
        Yp[(size_t)n * RP + r] = v;
        s2 += v * v;
#pragma unroll
        for (int k = 0; k < KC; ++k) xty[k] += X[(size_t)n * KC + k] * v;
    }
    float q = 0.f;
#pragma unroll
    for (int k = 0; k < KC; ++k) {
        float c = 0.f;
#pragma unroll
        for (int i = 0; i < KC; ++i) c += XtXinv[k * KC + i] * xty[i];
        q += xty[k] * c;
    }
    yty[r] = s2 - q;
}

// =====================================================================
// Kernel 4: WMMA GEMM   GY (LV x RP) = Gt^T (LV x NS) * Yp (NS x RP)
// TDM path: 2D-tile descriptors stream A (Gt slab, stride LV) and B
// (Yp slab, stride RP) chunks of 100 K-rows into double-buffered LDS
// via TENSOR_LOAD_TO_LDS; sync on TENSORcnt; fp32 16x16x4 WMMA inner
// loop consumes from LDS. One wave / 16x16 tile, grid 625 x 13.
// =====================================================================
#define CHUNK  100                       // K rows per TDM chunk (2000/100 = 20)
#define NCHUNK (NS / CHUNK)
#define SLABF  (CHUNK * 16)              // floats per slab
#define SLABB  (SLABF * 4)               // bytes per slab (6400)

#if __has_builtin(__builtin_amdgcn_tensor_load_to_lds) && \
    __has_builtin(__builtin_amdgcn_s_wait_tensorcnt)
#define HAVE_TDM 1
#else
#define HAVE_TDM 0
#endif

#if HAVE_TDM
// Build a D# (ISA ch.8) for a 2D tile: tile_rows x tile_cols of f32,
// row stride = stride_elems, destination LDS byte offset lds_off.
__device__ __forceinline__ void tdm_load_2d(const void* gaddr, unsigned lds_off,
                                            unsigned tile_rows, unsigned tile_cols,
                                            unsigned long long stride_elems,
                                            unsigned tensor_d0, unsigned tensor_d1) {
    unsigned long long ga = (unsigned long long)(uintptr_t)gaddr;
    v4u g0;
    g0.x = 1u;                                        // count=1, user descriptor
    g0.y = lds_off;                                   // lds_addr [63:32]
    g0.z = (unsigned)(ga & 0xFFFFFFFFu);              // global_addr lo
    g0.w = (unsigned)(ga >> 32) | (2u << 30);         // global_addr hi | type=2
    v8i g1;
    g1[0] = (int)(2u << 16);                          // data_size=4B, wg_mask=0
    g1[1] = (int)((tensor_d0 & 0xFFFFu) << 16);                                   // dim0[15:0]
    g1[2] = (int)(((tensor_d0 >> 16) & 0xFFFFu) | ((tensor_d1 & 0xFFFFu) << 16)); // dim0 hi | dim1 lo
    g1[3] = (int)(((tensor_d1 >> 16) & 0xFFFFu) | ((tile_cols & 0xFFFFu) << 16)); // dim1 hi | tile_dim0
    g1[4] = (int)(tile_rows & 0xFFFFu);               // tile_dim1 (tile_dim2=0)
    g1[5] = (int)(stride_elems & 0xFFFFFFFFull);      // tensor_dim0_stride lo
    g1[6] = (int)((stride_elems >> 32) & 0xFFFFull);  // stride hi (dim1_stride=0)
    g1[7] = 0;
    v4i z4 = {0, 0, 0, 0};
#if __clang_major__ >= 23
    v8i z8 = {0, 0, 0, 0, 0, 0, 0, 0};
    __builtin_amdgcn_tensor_load_to_lds(g0, g1, z4, z4, z8, 0);
#else
    __builtin_amdgcn_tensor_load_to_lds(g0, g1, z4, z4, 0);
#endif
}

__global__ void __launch_bounds__(32) gemm_wmma_tdm_kernel(const float* __restrict__ Gt,
                                                           const float* __restrict__ Yp,
                                                           float* __restrict__ GY) {
    extern __shared__ float smem[];      // [A0 | A1 | B0 | B1], SLABB each
    const int lane  = threadIdx.x & 31;
    const int m0    = blockIdx.x * 16;
    const int n0    = blockIdx.y * 16;
    const int mi    = lane & 15;
    const int khalf = (lane >> 4) << 1;  // 0 or 2

    auto issue = [&](int c) {
        unsigned buf = (unsigned)(c & 1);
        tdm_load_2d(Gt + (size_t)c * CHUNK * LV + m0, buf * SLABB,
                    CHUNK, 16, (unsigned long long)LV, LV, NS);
        tdm_load_2d(Yp + (size_t)c * CHUNK * RP + n0, (2u + buf) * SLABB,
                    CHUNK, 16, (unsigned long long)RP, RP, NS);
    };

    issue(0);
    v8f acc = {};
    for (int c = 0; c < NCHUNK; ++c) {
        if (c + 1 < NCHUNK) {
            issue(c + 1);                                   // prefetch next chunk
            __builtin_amdgcn_s_wait_tensorcnt((short)2);    // chunk c's 2 TDMs done
        } else {
            __builtin_amdgcn_s_wait_tensorcnt((short)0);
        }
        const float* lA = smem + (size_t)(c & 1) * SLABF;
        const float* lB = smem + (size_t)(2 + (c & 1)) * SLABF;
#pragma unroll 5
        for (int kk = 0; kk < CHUNK; kk += 4) {
            v2f a, b;
            a.x = lA[(kk + khalf) * 16 + mi];
            a.y = lA[(kk + khalf + 1) * 16 + mi];
            b.x = lB[(kk + khalf) * 16 + mi];
            b.y = lB[(kk + khalf + 1) * 16 + mi];
            acc = __builtin_amdgcn_wmma_f32_16x16x4_f32(false, a, false, b,
                                                        (short)0, acc, false, false);
        }
    }
    const int nc    = lane & 15;
    const int rbase = (lane >> 4) * 8;
#pragma unroll
    for (int v = 0; v < 8; ++v) {
        int m = v + rbase;
        GY[(size_t)(m0 + m) * RP + (n0 + nc)] = acc[v];
    }
}
#else
// Fallback: direct-global WMMA GEMM.
__global__ void __launch_bounds__(32) gemm_wmma_tdm_kernel(const float* __restrict__ Gt,
                                                           const float* __restrict__ Yp,
                                                           float* __restrict__ GY) {
    const int lane  = threadIdx.x & 31;
    const int m0    = blockIdx.x * 16;
    const int n0    = blockIdx.y * 16;
    const int mi    = lane & 15;
    const int khalf = (lane >> 4) << 1;
    const float* aptr = Gt + (size_t)khalf * LV + (m0 + mi);
    const float* bptr = Yp + (size_t)khalf * RP + (n0 + mi);
    v8f acc = {};
    for (int k = 0; k < NS; k += 8) {
        v2f a0, b0, a1, b1;
        a0.x = aptr[0];            a0.y = aptr[(size_t)LV];
        b0.x = bptr[0];            b0.y = bptr[(size_t)RP];
        a1.x = aptr[(size_t)4 * LV]; a1.y = aptr[(size_t)5 * LV];
        b1.x = bptr[(size_t)4 * RP]; b1.y = bptr[(size_t)5 * RP];
        __builtin_prefetch(aptr + (size_t)32 * LV, 0, 1);
        acc = __builtin_amdgcn_wmma_f32_16x16x4_f32(false, a0, false, b0,
                                                    (short)0, acc, false, false);
        acc = __builtin_amdgcn_wmma_f32_16x16x4_f32(false, a1, false, b1,
                                                    (short)0, acc, false, false);
        aptr += (size_t)8 * LV;
        bptr += (size_t)8 * RP;
    }
    const int nc    = lane & 15;
    const int rbase = (lane >> 4) * 8;
#pragma unroll
    for (int v = 0; v < 8; ++v)
        GY[(size_t)(m0 + v + rbase) * RP + (n0 + nc)] = acc[v];
}
#endif

// =====================================================================
// Kernel 5: per-permutation lead statistic + nominal p-value
// =====================================================================
__global__ void __launch_bounds__(256) perm_stat_kernel(const float* __restrict__ GY,
                                                        const float* __restrict__ gg,
                                                        const float* __restrict__ yty,
                                                        float* __restrict__ p_perm) {
    __shared__ float red[256];
    int r = blockIdx.x, t = threadIdx.x;
    float m = 0.f;
    for (int j = t; j < LV; j += 256) {
        float g = GY[(size_t)j * RP + r];
        m = fmaxf(m, g * g / gg[j]);
    }
    red[t] = m;
    __syncthreads();
    for (int s = 128; s > 0; s >>= 1) {
        if (t < s) red[t] = fmaxf(red[t], red[t + s]);
        __syncthreads();
    }
    if (t == 0) {
        float t2m = red[0];
        float ts2 = (float)DOF * t2m / (yty[r] - t2m);
        p_perm[r] = erfcf(sqrtf(ts2) * 0.70710678118654752f);
    }
}

// ---------------- device special functions for the Beta fit ----------
__device__ float digamma_(float x) {
    float r = 0.f;
    while (x < 6.f) { r -= 1.f / x; x += 1.f; }
    float inv = 1.f / x, i2 = inv * inv;
    return r + logf(x) - 0.5f * inv
           - i2 * (1.f / 12.f - i2 * (1.f / 120.f - i2 * (1.f / 252.f)));
}
__device__ float trigamma_(float x) {
    float r = 0.f;
    while (x < 6.f) { r += 1.f / (x * x); x += 1.f; }
    float inv = 1.f / x, i2 = inv * inv;
    return r + inv * (1.f + inv * (0.5f + inv * (1.f / 6.f - i2 * (1.f / 30.f - i2 * (1.f / 42.f)))));
}
__device__ float tetragamma_(float x) {
    float r = 0.f;
    while (x < 6.f) { r -= 2.f / (x * x * x); x += 1.f; }
    float inv = 1.f / x, i2 = inv * inv;
    return r - i2 * (1.f + inv * (1.f + inv * (0.5f - i2 * (1.f / 6.f - i2 * 0.3f))));
}
__device__ float betacf_(float a, float b, float x) {
    const float FPMIN = 1e-30f;
    float qab = a + b, qap = a + 1.f, qam = a - 1.f;
    float c = 1.f, d = 1.f - qab * x / qap;
    if (fabsf(d) < FPMIN) d = FPMIN;
    d = 1.f / d;
    float h = d;
    for (int m = 1; m <= 100; ++m) {
        float m2 = 2.f * m;
        float aa = m * (b - m) * x / ((qam + m2) * (a + m2));
        d = 1.f + aa * d; if (fabsf(d) < FPMIN) d = FPMIN;
        c = 1.f + aa / c; if (fabsf(c) < FPMIN) c = FPMIN;
        d = 1.f / d; h *= d * c;
        aa = -(a + m) * (qab + m) * x / ((a + m2) * (qap + m2));
        d = 1.f + aa * d; if (fabsf(d) < FPMIN) d = FPMIN;
        c = 1.f + aa / c; if (fabsf(c) < FPMIN) c = FPMIN;
        d = 1.f / d;
        float del = d * c;
        h *= del;
        if (fabsf(del - 1.f) < 3e-7f) break;
    }
    return h;
}
__device__ float betainc_(float a, float b, float x) {
    if (x <= 0.f) return 0.f;
    if (x >= 1.f) return 1.f;
    float bt = expf(lgammaf(a + b) - lgammaf(a) - lgammaf(b)
                    + a * logf(x) + b * logf(1.f - x));
    if (x < (a + 1.f) / (a + b + 2.f)) return bt * betacf_(a, b, x) / a;
    return 1.f - bt * betacf_(b, a, 1.f - x) / b;
}

// =====================================================================
// Kernel 6: moments init + Newton-Raphson with Christoffel correction
// (faithful port of _infer_beta) + betainc of min(obs_p). One block.
// =====================================================================
__global__ void __launch_bounds__(256) finalize_kernel(const float* __restrict__ p_perm,
                                                       const float* __restrict__ obs_p,
                                                       float* __restrict__ out) {
    __shared__ float sp[256], sp2[256], slp[256], sl1p[256], smin[256];
    int t = threadIdx.x;
    float p = 0.f, p2 = 0.f, lp = 0.f, l1p = 0.f, mn = 3.4e38f;
    if (t < RNUM) {
        float pp = p_perm[t];
        p = pp; p2 = pp * pp;
        lp  = logf(fmaxf(pp, 1e-37f));
        l1p = logf(fmaxf(1.f - pp, 1e-37f));
    }
    for (int j = t; j < LV; j += 256) mn = fminf(mn, obs_p[j]);
    sp[t] = p; sp2[t] = p2; slp[t] = lp; sl1p[t] = l1p; smin[t] = mn;
    __syncthreads();
    for (int s = 128; s > 0; s >>= 1) {
        if (t < s) {
            sp[t] += sp[t + s]; sp2[t] += sp2[t + s];
            slp[t] += slp[t + s]; sl1p[t] += sl1p[t + s];
            smin[t] = fminf(smin[t], smin[t + s]);
        }
        __syncthreads();
    }
    if (t != 0) return;

    const float nobs = (float)RNUM;
    float mean = sp[0] / nobs;
    float var  = sp2[0] / nobs - mean * mean;
    float slogp = slp[0], slog1p = sl1p[0], minp = smin[0];

    float kk = mean * (mean * (1.f - mean) / var - 1.f);
    float nn = kk * (1.f / mean - 1.f);

    float old_lik = 10000.f, diff = 1000.f;
    int it = 0;
    for (int step = 0; step <= MAXIT_BETA; ++step) {
        bool active = (fabsf(diff) > TOLF) && (it <= MAXIT_BETA);
        float pg1k = trigamma_(kk), pg1n = trigamma_(nn), pg1kn = trigamma_(kk + nn);
        float pg2k = tetragamma_(kk), pg2n = tetragamma_(nn), pg2kn = tetragamma_(kk + nn);
        float dg_kn = digamma_(kk + nn);
        float s0 = slogp  + nobs * (dg_kn - digamma_(kk));
        float s1 = slog1p + nobs * (dg_kn - digamma_(nn));
        float i_kn = -pg1kn;
        float i_k  = pg1k + i_kn;
        float i_n  = pg1n + i_kn;
        float I00 = -nobs * i_k, I01 = -nobs * i_kn, I11 = -nobs * i_n;
        float det = I00 * I11 - I01 * I01;
        float d0 = (I11 * s0 - I01 * s1) / det;
        float d1 = (I00 * s1 - I01 * s0) / det;
        float i_kkn = pg1n * pg2kn;
        float g_k   = -pg1n * pg2k + i_kkn + pg1kn * pg2k;
        float i_knn = i_kkn - pg1kn * pg2n;
        float i_nnk = pg1k * pg2kn;
        float i_nkk = i_nnk - pg1kn * pg2k;
        float g_n   = -pg1k * pg2n + i_nnk + pg1kn * pg2n;
        float scale = -pg1k * pg1n + (pg1k + pg1n) * pg1kn;
        float inv2s = 0.5f / scale;
        float adj0 = inv2s * (g_k  * d0 * d0 + 2.f * i_kkn * d0 * d1 + i_knn * d1 * d1);
        float adj1 = inv2s * (i_nkk * d0 * d0 + 2.f * i_nnk * d0 * d1 + g_n  * d1 * d1);
        float nk  = kk - d0 - 0.5f * adj0;
        float nnv = nn - d1 - 0.5f * adj1;
        float new_lik = (nk - 1.f) * slogp + (nnv - 1.f) * slog1p
                        + nobs * (lgammaf(nk + nnv) - lgammaf(nk) - lgammaf(nnv));
        float new_diff = old_lik - new_lik;
        if (active) {
            old_lik = new_lik; diff = new_diff; it += 1; kk = nk; nn = nnv;
        }
    }
    float converged = ((fabsf(diff) < TOLF) && (it <= MAXIT_BETA)) ? 1.f : 0.f;

    out[0] = betainc_(kk, nn, minp);   // adj_p
    out[1] = kk;
    out[2] = nn;
    out[3] = converged;
}

// =====================================================================
// Host launcher. Workspace: ~91 MB (Gt dominates at 80 MB; L2-resident
// on MI455X's 192 MB L2 for the TDM+WMMA GEMM pass).
// =====================================================================
extern "C" void kernel_launch(void* const* d_in, const int* in_sizes, int n_in,
                              void* d_out, int out_size, void* d_ws, size_t ws_size,
                              hipStream_t stream) {
    const float* X          = (const float*)d_in[0];
    const float* y          = (const float*)d_in[1];
    const float* G          = (const float*)d_in[2];
    const float* obs_p      = (const float*)d_in[3];
    const float* offset_eta = (const float*)d_in[4];
    const int*   seed       = (const int*)d_in[5];
    float* out = (float*)d_out;

    char* ws = (char*)d_ws;
    size_t off = 0;
    auto take = [&](size_t nbytes) {
        size_t r = off;
        off += (nbytes + 255) & ~(size_t)255;
        return r;
    };
    float* XtXinv = (float*)(ws + take(KC * KC * sizeof(float)));
    float* Gt     = (float*)(ws + take((size_t)NS * LV * sizeof(float)));
    float* gg     = (float*)(ws + take((size_t)LV * sizeof(float)));
    float* Yp     = (float*)(ws + take((size_t)NS * RP * sizeof(float)));
    float* yty    = (float*)(ws + take((size_t)RP * sizeof(float)));
    float* GY     = (float*)(ws + take((size_t)LV * RP * sizeof(float)));
    float* pperm  = (float*)(ws + take((size_t)RP * sizeof(float)));

    xtx_inv_kernel<<<1, 128, 0, stream>>>(X, XtXinv);
    residualize_kernel<<<(LV + 255) / 256, 256, 0, stream>>>(X, G, XtXinv, Gt, gg);
    perm_kernel<<<1, 256, 0, stream>>>(y, offset_eta, X, XtXinv, seed, Yp, yty);

    dim3 gemm_grid(LV / 16, RP / 16);   // 625 x 13 tiles, one wave each
#if HAVE_TDM
    gemm_wmma_tdm_kernel<<<gemm_grid, 32, 4 * SLABB, stream>>>(Gt, Yp, GY);
#else
    gemm_wmma_tdm_kernel<<<gemm_grid, 32, 0, stream>>>(Gt, Yp, GY);
#endif

    perm_stat_kernel<<<RNUM, 256, 0, stream>>>(GY, gg, yty, pperm);
    finalize_kernel<<<1, 256, 0, stream>>>(pperm, obs_p, out);
}